// TokenEmbedding_30614526886303
// MI455X (gfx1250) — compile-verified
//
#include <hip/hip_runtime.h>

// TokenEmbedding: out[b,s,:] = W[:, tokens[b,s]]
//   tokens: [8, 4096] int32        (in_sizes[0] = 32768)
//   W:      [EMB=1024, VOCAB=32000] float32 (row-major)
//   out:    [8, 4096, 1024] float32
//
// Pure gather, 0 FLOPs -> memory-roofline problem (~256 MiB total HBM
// traffic, ~11 us at 23.3 TB/s). WMMA is deliberately NOT used: the one-hot
// matmul formulation would do 32000x more work than the gather.
//
// MI455X-specific reasoning:
//  * W (131 MB) < global L2 (192 MB): scattered column reads become L2 hits
//    after first touch (each W line reused ~33x across random tokens), so we
//    keep W loads at default (RT) temporal policy.
//  * The 128 MiB output is write-once/stream: use non-temporal stores
//    (global_store_b128 with NT temporal hint) so the output stream does not
//    evict W from L2.
//  * wave32: 256-thread block = 8 waves; thread t writes elements 4t..4t+3
//    -> one coalesced B128 store per thread (512 B contiguous per wave),
//    4 independent B32 loads in flight per thread for MLP.
//  * token id is uniform per block -> scalar s_load.

#define VOCAB 32000
#define EMB   1024

// Clang native vector type: accepted by __builtin_nontemporal_store
// (HIP's float4 struct is not).
typedef float v4f __attribute__((ext_vector_type(4)));

__global__ __launch_bounds__(256)
void TokenEmbedding_gather_kernel(const int*   __restrict__ tokens,
                                  const float* __restrict__ W,
                                  float*       __restrict__ out,
                                  int ntok)
{
    const int t = blockIdx.x;          // one block per token
    if (t >= ntok) return;

    const int tid = threadIdx.x;       // 0..255
    const int e   = tid << 2;          // this thread's 4 embedding rows

    const int tok = tokens[t];         // uniform -> scalar load

    // Column gather: W[e + j][tok], j = 0..3. Stride between elements is
    // VOCAB*4 = 128 KB -> one cache line per lane; L2-resident after warmup.
    const float* __restrict__ col = W + (size_t)tok;
    v4f v;
    v.x = col[(size_t)(e + 0) * VOCAB];
    v.y = col[(size_t)(e + 1) * VOCAB];
    v.z = col[(size_t)(e + 2) * VOCAB];
    v.w = col[(size_t)(e + 3) * VOCAB];

    // Streaming (non-temporal) 16-byte store: coalesced across the wave,
    // bypasses L2 retention so W stays resident.
    v4f* dst = (v4f*)(out + (size_t)t * EMB + e);
    __builtin_nontemporal_store(v, dst);
}

extern "C" void kernel_launch(void* const* d_in, const int* in_sizes, int n_in,
                              void* d_out, int out_size, void* d_ws, size_t ws_size,
                              hipStream_t stream)
{
    const int*   tokens = (const int*)d_in[0];   // int32 tokens
    const float* W      = (const float*)d_in[1]; // [EMB, VOCAB] fp32
    float*       out    = (float*)d_out;         // [ntok, EMB] fp32

    const int ntok = in_sizes[0];                // 8 * 4096 = 32768

    dim3 block(256);                             // 8 wave32s; covers EMB with x4 vec
    dim3 grid(ntok);                             // one block per token

    TokenEmbedding_gather_kernel<<<grid, block, 0, stream>>>(tokens, W, out, ntok);
}